// TableGCN_71038759076318
// MI455X (gfx1250) — compile-verified
//
#include <hip/hip_runtime.h>

typedef float v2f __attribute__((ext_vector_type(2)));
typedef float v8f __attribute__((ext_vector_type(8)));

#define DFEAT 128
#define NGRAPHS 16

// ---------------- degree / normalization ----------------

__global__ void k_init_deg(float* __restrict__ deg, int n) {
    int i = blockIdx.x * blockDim.x + threadIdx.x;
    if (i < n) deg[i] = 1.0f;  // self-loop contributes 1
}

__global__ void k_count_deg(const int* __restrict__ dst, float* __restrict__ deg, int e) {
    int i = blockIdx.x * blockDim.x + threadIdx.x;
    if (i < e) unsafeAtomicAdd(&deg[dst[i]], 1.0f);
}

__global__ void k_dis(float* __restrict__ deg, int n) {
    int i = blockIdx.x * blockDim.x + threadIdx.x;
    if (i < n) deg[i] = rsqrtf(deg[i]);   // deg >= 1 always (self loops)
}

// ---------------- fp32 WMMA GEMM: C[n x 128] = A[n x 128] * W[128 x 128] ----------------
// One wave computes one 16x16 tile of C. blockDim=256 (8 waves), wave w -> col tile w.
// Uses V_WMMA_F32_16X16X4_F32 (full fp32, K=4), 32 accumulating WMMAs per tile.

__global__ __launch_bounds__(256) void k_gemm_wmma(const float* __restrict__ A,
                                                   const float* __restrict__ W,
                                                   float* __restrict__ C,
                                                   int nrows) {
    const int wave = threadIdx.x >> 5;   // 0..7 -> column tile
    const int lane = threadIdx.x & 31;
    const int row0 = blockIdx.x * 16;
    const int col0 = wave * 16;
    if (row0 >= nrows) return;

    const int mr = lane & 15;            // M (for A) / N (for B,C)
    const int kh = lane >> 4;            // which K-half within a K=4 step

    v8f acc = {};
    const float* arow = A + (size_t)(row0 + mr) * DFEAT;

#pragma unroll
    for (int k0 = 0; k0 < DFEAT; k0 += 4) {
        const int ka = k0 + 2 * kh;
        // A fragment (16x4): lane m holds A[m][ka], A[m][ka+1] in 2 VGPRs
        v2f a;
        a.x = arow[ka];
        a.y = arow[ka + 1];
        // B fragment (4x16): lane n holds B[ka][n], B[ka+1][n]
        v2f b;
        b.x = W[(size_t)ka * DFEAT + col0 + mr];
        b.y = W[(size_t)(ka + 1) * DFEAT + col0 + mr];
        acc = __builtin_amdgcn_wmma_f32_16x16x4_f32(
            /*neg_a=*/false, a, /*neg_b=*/false, b,
            /*c_mod=*/(short)0, acc, /*reuse_a=*/false, /*reuse_b=*/false);
    }

    // C layout: VGPR g, lanes 0-15 -> row g; lanes 16-31 -> row g+8; col = lane&15
#pragma unroll
    for (int g = 0; g < 8; ++g) {
        C[(size_t)(row0 + g + 8 * kh) * DFEAT + col0 + mr] = acc[g];
    }
}

// ---------------- edge scatter: agg[dst] += h[src] * dis[src]*dis[dst] ----------------
// 1 wave per edge; lane handles 4 floats (32*4 = 128).

__global__ __launch_bounds__(256) void k_scatter(const int* __restrict__ src,
                                                 const int* __restrict__ dst,
                                                 const float* __restrict__ dis,
                                                 const float* __restrict__ h,
                                                 float* __restrict__ agg,
                                                 int e) {
    const int wave = threadIdx.x >> 5;
    const int lane = threadIdx.x & 31;
    const int eidx = blockIdx.x * 8 + wave;
    if (eidx >= e) return;
    const int s = src[eidx];
    const int d = dst[eidx];
    const float nrm = dis[s] * dis[d];
    const float4 v = *(const float4*)(h + (size_t)s * DFEAT + lane * 4);
    float* ap = agg + (size_t)d * DFEAT + lane * 4;
    unsafeAtomicAdd(ap + 0, v.x * nrm);
    unsafeAtomicAdd(ap + 1, v.y * nrm);
    unsafeAtomicAdd(ap + 2, v.z * nrm);
    unsafeAtomicAdd(ap + 3, v.w * nrm);
}

// ---------------- combine: self-loop term + bias (+ optional relu) ----------------

__global__ void k_combine_relu(const float* __restrict__ agg, float* __restrict__ h,
                               const float* __restrict__ dis, const float* __restrict__ bias,
                               int n) {
    const int i = blockIdx.x * blockDim.x + threadIdx.x;
    const int node = i >> 7, col = i & 127;
    if (node < n) {
        const float ds = dis[node];
        const float v = agg[i] + h[i] * ds * ds + bias[col];
        h[i] = fmaxf(v, 0.0f);
    }
}

__global__ void k_combine_out(float* __restrict__ outp /* holds agg2, in-place */,
                              const float* __restrict__ h2,
                              const float* __restrict__ dis, const float* __restrict__ bias,
                              int n) {
    const int i = blockIdx.x * blockDim.x + threadIdx.x;
    const int node = i >> 7, col = i & 127;
    if (node < n) {
        const float ds = dis[node];
        outp[i] = outp[i] + h2[i] * ds * ds + bias[col];
    }
}

// ---------------- global mean pool ----------------
// 128 threads/block; thread t owns feature column t in an LDS [16][128] accumulator.

__global__ __launch_bounds__(128) void k_pool(const float* __restrict__ outp,
                                              const int* __restrict__ batch,
                                              float* __restrict__ gsum,
                                              float* __restrict__ gcnt,
                                              int n, int chunk) {
    __shared__ float lsum[NGRAPHS * DFEAT];
    __shared__ float lcnt[NGRAPHS];
    const int t = threadIdx.x;
    for (int j = t; j < NGRAPHS * DFEAT; j += 128) lsum[j] = 0.0f;
    if (t < NGRAPHS) lcnt[t] = 0.0f;
    __syncthreads();

    const int i0 = blockIdx.x * chunk;
    int i1 = i0 + chunk;
    if (i1 > n) i1 = n;
    for (int i = i0; i < i1; ++i) {
        const int b = batch[i];
        lsum[b * DFEAT + t] += outp[(size_t)i * DFEAT + t];
        if (t == 0) lcnt[b] += 1.0f;
    }
    __syncthreads();

#pragma unroll
    for (int b = 0; b < NGRAPHS; ++b)
        unsafeAtomicAdd(&gsum[b * DFEAT + t], lsum[b * DFEAT + t]);
    if (t < NGRAPHS) unsafeAtomicAdd(&gcnt[t], lcnt[t]);
}

__global__ void k_pool_div(float* __restrict__ g, const float* __restrict__ gcnt) {
    const int i = blockIdx.x * blockDim.x + threadIdx.x;   // 2048 elems
    if (i < NGRAPHS * DFEAT) {
        const int b = i >> 7;
        g[i] = g[i] / fmaxf(gcnt[b], 1.0f);
    }
}

// ---------------- host orchestration ----------------

extern "C" void kernel_launch(void* const* d_in, const int* in_sizes, int n_in,
                              void* d_out, int out_size, void* d_ws, size_t ws_size,
                              hipStream_t stream) {
    const float* x    = (const float*)d_in[0];
    const int*   ei   = (const int*)d_in[1];   // [2, E] flat
    const int*   batch= (const int*)d_in[2];
    const float* W1   = (const float*)d_in[3];
    const float* b1   = (const float*)d_in[4];
    const float* W2   = (const float*)d_in[5];
    const float* b2   = (const float*)d_in[6];

    const int N = in_sizes[0] / DFEAT;
    const int E = in_sizes[1] / 2;
    const int* esrc = ei;
    const int* edst = ei + E;

    // workspace layout
    char* ws = (char*)d_ws;
    float* dis  = (float*)ws;                        // N floats (deg -> dis in place)
    size_t off  = (size_t)N * sizeof(float);
    off = (off + 255) & ~(size_t)255;
    float* gcnt = (float*)(ws + off);                // 16 floats
    off += 256;
    float* bufA = (float*)(ws + off);                // N*128 floats
    off += (size_t)N * DFEAT * sizeof(float);
    float* bufB = (float*)(ws + off);                // N*128 floats
    (void)ws_size;

    float* out_graph = (float*)d_out;                // [16,128]
    float* out_nodes = (float*)d_out + NGRAPHS * DFEAT;  // [N,128]

    const int TB = 256;

    // 1) degree + normalization
    k_init_deg<<<(N + TB - 1) / TB, TB, 0, stream>>>(dis, N);
    k_count_deg<<<(E + TB - 1) / TB, TB, 0, stream>>>(edst, dis, E);
    k_dis<<<(N + TB - 1) / TB, TB, 0, stream>>>(dis, N);

    // 2) layer 1: h1 = x @ W1 (WMMA fp32)
    k_gemm_wmma<<<N / 16, TB, 0, stream>>>(x, W1, bufA, N);
    hipMemsetAsync(bufB, 0, (size_t)N * DFEAT * sizeof(float), stream);
    k_scatter<<<(E + 7) / 8, TB, 0, stream>>>(esrc, edst, dis, bufA, bufB, E);
    // h = relu(agg + h1*dis^2 + b1)   (bufA becomes h, in place)
    k_combine_relu<<<((size_t)N * DFEAT + TB - 1) / TB, TB, 0, stream>>>(bufB, bufA, dis, b1, N);

    // 3) layer 2: h2 = h @ W2
    k_gemm_wmma<<<N / 16, TB, 0, stream>>>(bufA, W2, bufB, N);
    hipMemsetAsync(d_out, 0, (size_t)out_size * sizeof(float), stream);
    k_scatter<<<(E + 7) / 8, TB, 0, stream>>>(esrc, edst, dis, bufB, out_nodes, E);
    k_combine_out<<<((size_t)N * DFEAT + TB - 1) / TB, TB, 0, stream>>>(out_nodes, bufB, dis, b2, N);

    // 4) global mean pool
    hipMemsetAsync(gcnt, 0, NGRAPHS * sizeof(float), stream);
    const int nblocks = 400;
    const int chunk = (N + nblocks - 1) / nblocks;
    k_pool<<<nblocks, 128, 0, stream>>>(out_nodes, batch, out_graph, gcnt, N, chunk);
    k_pool_div<<<(NGRAPHS * DFEAT + TB - 1) / TB, TB, 0, stream>>>(out_graph, gcnt);
}